// MLDEL_6_52269751992451
// MI455X (gfx1250) — compile-verified
//
#include <hip/hip_runtime.h>
#include <math.h>

#define EPS_H  1e-7f
#define MINN_H 1e-15f
#define MAXN_H 1e6f

typedef float v2f __attribute__((ext_vector_type(2)));
typedef float v8f __attribute__((ext_vector_type(8)));

// ---------------- wave32 helpers ----------------
__device__ __forceinline__ float wave_sum32(float v) {
#pragma unroll
  for (int off = 16; off > 0; off >>= 1) v += __shfl_xor(v, off, 32);
  return v;
}
__device__ __forceinline__ float dot4(float4 a, float4 b) {
  return a.x*b.x + a.y*b.y + a.z*b.z + a.w*b.w;
}
__device__ __forceinline__ float4 f4s(float4 a, float s) {
  return make_float4(a.x*s, a.y*s, a.z*s, a.w*s);
}
__device__ __forceinline__ float4 f4a(float4 a, float4 b) {
  return make_float4(a.x+b.x, a.y+b.y, a.z+b.z, a.w+b.w);
}
__device__ __forceinline__ float4 f4axpy(float s, float4 a, float4 b) { // s*a + b
  return make_float4(fmaf(s,a.x,b.x), fmaf(s,a.y,b.y), fmaf(s,a.z,b.z), fmaf(s,a.w,b.w));
}
__device__ __forceinline__ float arcosh_(float x) {
  x = fmaxf(x, 1.0f + EPS_H);
  return logf(x + sqrtf(x*x - 1.0f));
}

// logmap0: hyperboloid point fragment -> tangent fragment (comp0 = 0). One wave = one 128-d row.
__device__ __forceinline__ float4 logmap0_frag(float4 x, int lane, float sqrtK) {
  float x0 = __shfl(x.x, 0, 32);
  float4 y = x;
  if (lane == 0) y.x = 0.0f;
  float yn = fmaxf(sqrtf(wave_sum32(dot4(y, y))), MINN_H);
  float t = sqrtK * arcosh_(x0 / sqrtK) / yn;
  return f4s(y, t);
}

// expmap0 + proj: tangent fragment (comp0 ignored) -> hyperboloid point fragment.
__device__ __forceinline__ float4 expmap0_frag(float4 u, int lane, float sqrtK, float K) {
  float4 v = u;
  if (lane == 0) v.x = 0.0f;
  float xn = fmaxf(sqrtf(wave_sum32(dot4(v, v))), MINN_H);
  float theta = xn / sqrtK;
  float s = sqrtK * sinhf(theta) / xn;
  float4 r = f4s(v, s);          // comp0 stays 0 on lane0
  float sn = wave_sum32(dot4(r, r));
  if (lane == 0) r.x = sqrtf(K + sn);   // proj
  return r;
}

// ---------------- scalar / bias prep ----------------
__global__ void k_scalars(const float* __restrict__ raw_c, float* __restrict__ sc) {
  if (threadIdx.x == 0 && blockIdx.x == 0) {
    float x = raw_c[0];
    float c = ((x > 20.0f) ? x : log1pf(expf(x))) + 1e-5f;   // softplus + 1e-5
    sc[0] = c;
    sc[1] = sqrtf(1.0f / c);
    sc[2] = 1.0f / c;
    sc[3] = 0.0f;   // loss accumulator (reset every call)
  }
}

// ub = logmap0(proj(expmap0(proj_tan0(b))))  -- one wave, 128-dim bias
__global__ void k_bias_prep(const float* __restrict__ b, float* __restrict__ ub,
                            const float* __restrict__ sc) {
  int lane = threadIdx.x;
  float sqrtK = sc[1], K = sc[2];
  float4 u = ((const float4*)b)[lane];
  if (lane == 0) u.x = 0.0f;                         // proj_tan0
  float4 y  = expmap0_frag(u, lane, sqrtK, K);       // hyperbolic bias point
  float4 uo = logmap0_frag(y, lane, sqrtK);
  ((float4*)ub)[lane] = uo;
}

// ---------------- wave-per-row elementwise kernels ----------------
__global__ __launch_bounds__(256) void k_logmap0_rows(const float* __restrict__ X,
                                                      float* __restrict__ U,
                                                      const float* __restrict__ sc, int M) {
  int lane = threadIdx.x & 31, wave = threadIdx.x >> 5;
  int row = blockIdx.x * 8 + wave;
  if (row >= M) return;
  float sqrtK = sc[1];
  float4 x = ((const float4*)(X + (size_t)row * 128))[lane];
  ((float4*)(U + (size_t)row * 128))[lane] = logmap0_frag(x, lane, sqrtK);
}

__global__ __launch_bounds__(256) void k_expmap0_rows(const float* __restrict__ V,
                                                      float* __restrict__ OUT,
                                                      const float* __restrict__ sc, int M) {
  int lane = threadIdx.x & 31, wave = threadIdx.x >> 5;
  int row = blockIdx.x * 8 + wave;
  if (row >= M) return;
  float sqrtK = sc[1], K = sc[2];
  float4 v = ((const float4*)(V + (size_t)row * 128))[lane];
  ((float4*)(OUT + (size_t)row * 128))[lane] = expmap0_frag(v, lane, sqrtK, K);
}

// x = mobius_add(x, bias_point)  with ub = logmap0(bias_point) precomputed (in-place)
__global__ __launch_bounds__(256) void k_madd_bias(float* X, const float* __restrict__ ub,
                                                   const float* __restrict__ sc, int M) {
  int lane = threadIdx.x & 31, wave = threadIdx.x >> 5;
  int row = blockIdx.x * 8 + wave;
  if (row >= M) return;
  float sqrtK = sc[1], K = sc[2];
  float4* xp = (float4*)(X + (size_t)row * 128);
  float4 x = xp[lane];
  float4 u = ((const float4*)ub)[lane];
  float x0 = __shfl(x.x, 0, 32);
  // ptransp0
  float4 d = x; if (lane == 0) d.x = 0.0f;
  float yn = fmaxf(sqrtf(wave_sum32(dot4(d, d))), MINN_H);
  float4 yhat = f4s(d, 1.0f / yn);
  float4 vv = f4s(yhat, sqrtK - x0);
  if (lane == 0) vv.x = -yn;
  float alpha = wave_sum32(dot4(yhat, u)) / sqrtK;     // comp0 terms are 0
  float4 w = f4axpy(-alpha, vv, u);                    // u - alpha*v
  // proj_tan
  float4 xs = x; if (lane == 0) xs.x = 0.0f;
  float ux = wave_sum32(dot4(xs, w));
  if (lane == 0) w.x = ux / fmaxf(x0, EPS_H);
  // expmap(w, x)
  float w0 = __shfl(w.x, 0, 32);
  float mdot = wave_sum32(dot4(w, w)) - 2.0f * w0 * w0;
  float normu = fminf(sqrtf(fmaxf(mdot, EPS_H)), MAXN_H);
  float theta = fmaxf(normu / sqrtK, MINN_H);
  float ch = coshf(theta);
  float sh = sinhf(theta) / theta;
  float4 res = f4a(f4s(x, ch), f4s(w, sh));
  float4 rs = res; if (lane == 0) rs.x = 0.0f;
  float sn = wave_sum32(dot4(rs, rs));
  if (lane == 0) res.x = sqrtf(K + sn);                // proj
  xp[lane] = res;
}

// out = expmap0(s * logmap0(x)),  s = n[row] (or 1-n[row])
__global__ __launch_bounds__(256) void k_matvec0(const float* __restrict__ X,
                                                 const float* __restrict__ nv,
                                                 float* __restrict__ OUT, int oneMinus,
                                                 const float* __restrict__ sc, int M) {
  int lane = threadIdx.x & 31, wave = threadIdx.x >> 5;
  int row = blockIdx.x * 8 + wave;
  if (row >= M) return;
  float sqrtK = sc[1], K = sc[2];
  float s = nv[row]; if (oneMinus) s = 1.0f - s;
  float4 x = ((const float4*)(X + (size_t)row * 128))[lane];
  float4 u = f4s(logmap0_frag(x, lane, sqrtK), s);
  ((float4*)(OUT + (size_t)row * 128))[lane] = expmap0_frag(u, lane, sqrtK, K);
}

// x = expmap0(logmap0(x) + logmap0(y))  (in-place on x)
__global__ __launch_bounds__(256) void k_madd0(float* X, const float* __restrict__ Y,
                                               const float* __restrict__ sc, int M) {
  int lane = threadIdx.x & 31, wave = threadIdx.x >> 5;
  int row = blockIdx.x * 8 + wave;
  if (row >= M) return;
  float sqrtK = sc[1], K = sc[2];
  float4* xp = (float4*)(X + (size_t)row * 128);
  float4 x = xp[lane];
  float4 y = ((const float4*)(Y + (size_t)row * 128))[lane];
  float4 u = f4a(logmap0_frag(x, lane, sqrtK), logmap0_frag(y, lane, sqrtK));
  xp[lane] = expmap0_frag(u, lane, sqrtK, K);
}

__global__ void k_zero(float* p, long n) {
  long i = (long)blockIdx.x * blockDim.x + threadIdx.x;
  long stride = (long)gridDim.x * blockDim.x;
  for (; i < n; i += stride) p[i] = 0.0f;
}

// segment_sum: agg[rows[e]] += vals[e] * g[cols[e]]  (one wave per edge; agg is L2-resident)
__global__ __launch_bounds__(256) void k_scatter(const int* __restrict__ rows,
                                                 const int* __restrict__ cols,
                                                 const float* __restrict__ vals,
                                                 const float* __restrict__ g,
                                                 float* __restrict__ agg, int E) {
  int lane = threadIdx.x & 31, wave = threadIdx.x >> 5;
  int e = blockIdx.x * 8 + wave;
  if (e >= E) return;
  // prefetch the gather row of the edge 8 ahead (hides index->row dependent latency)
  int e2 = e + 8;
  if (e2 < E) {
    int c2 = cols[e2];
    __builtin_prefetch(g + (size_t)c2 * 128 + lane * 4, 0, 1);
  }
  int cI = cols[e], rI = rows[e];
  float w = vals[e];
  float4 gv = ((const float4*)(g + (size_t)cI * 128))[lane];
  float* dst = agg + (size_t)rI * 128 + lane * 4;
  atomicAdd(dst + 0, w * gv.x);
  atomicAdd(dst + 1, w * gv.y);
  atomicAdd(dst + 2, w * gv.z);
  atomicAdd(dst + 3, w * gv.w);
}

// selcat[b][l] = logmap0(x2[idx[b*L+l]])
__global__ __launch_bounds__(256) void k_gather_logmap(const float* __restrict__ X2,
                                                       const int* __restrict__ idx,
                                                       float* __restrict__ selc,
                                                       const float* __restrict__ sc,
                                                       int total, int L) {
  int lane = threadIdx.x & 31, wave = threadIdx.x >> 5;
  int w = blockIdx.x * 8 + wave;
  if (w >= total) return;
  float sqrtK = sc[1];
  int node = idx[w];
  int b = w / L, l = w % L;
  float4 x = ((const float4*)(X2 + (size_t)node * 128))[lane];
  ((float4*)(selc + ((size_t)b * (L + 1) + l) * 128))[lane] = logmap0_frag(x, lane, sqrtK);
}

// selcat[b][L] = sum_l selcat[b][l]
__global__ __launch_bounds__(256) void k_batch_sum(float* selc, int B, int L) {
  int lane = threadIdx.x & 31, wave = threadIdx.x >> 5;
  int b = blockIdx.x * 8 + wave;
  if (b >= B) return;
  float4 acc = make_float4(0, 0, 0, 0);
  for (int l = 0; l < L; ++l)
    acc = f4a(acc, ((const float4*)(selc + ((size_t)b * (L + 1) + l) * 128))[lane]);
  ((float4*)(selc + ((size_t)b * (L + 1) + L) * 128))[lane] = acc;
}

// ---------------- WMMA fp32 GEMM: C[M x NCOL] = A[M x 128] * W[128 x NCOL] ----------------
// 256 threads = 8 waves; each wave owns a 16-row strip; A tile + W staged in dynamic LDS.
// W is stored K-pair-swizzled so each B fragment is a single aligned ds_load_b64:
//   ldsW[((k>>1)*NCOL + n)*2 + (k&1)] = W[k][n]
template <int NCOL, bool RELU>
__global__ __launch_bounds__(256) void k_gemm128(const float* __restrict__ A,
                                                 const float* __restrict__ W,
                                                 float* __restrict__ C, int M) {
  extern __shared__ float smem[];
  float* ldsW = smem;               // [64][NCOL][2] pair-swizzled
  float* ldsA = smem + 128 * NCOL;  // [128][128]
  const int tid = threadIdx.x, wave = tid >> 5, lane = tid & 31;
  const int rowBlock = blockIdx.x * 128;
  for (int i = tid; i < 128 * NCOL; i += 256) {
    int k = i / NCOL, n = i % NCOL;           // NCOL is a power of two -> shifts
    ldsW[(((k >> 1) * NCOL + n) << 1) | (k & 1)] = W[i];
  }
  for (int i = tid; i < 128 * 128; i += 256) {
    int r = i >> 7, cix = i & 127;
    int gr = rowBlock + r; if (gr >= M) gr = M - 1;
    ldsA[i] = A[(size_t)gr * 128 + cix];
  }
  __syncthreads();
  const int mrow = lane & 15;      // N index within 16-tile / M index for A
  const int khalf = lane >> 4;     // which K pair (0 -> K0,K1 ; 1 -> K2,K3)
  const v2f* ap = (const v2f*)(ldsA + (wave * 16 + mrow) * 128);
  const v2f* wp = (const v2f*)ldsW;
  for (int j = 0; j < NCOL / 16; ++j) {
    v8f acc = {};
    for (int k = 0; k < 128; k += 4) {
      int p = (k >> 1) + khalf;                  // K-pair index
      v2f a = ap[p];                             // A[m][2p], A[m][2p+1]
      v2f b = wp[p * NCOL + j * 16 + mrow];      // W[2p][n], W[2p+1][n]
      acc = __builtin_amdgcn_wmma_f32_16x16x4_f32(false, a, false, b, (short)0, acc,
                                                  false, false);
    }
#pragma unroll
    for (int rr = 0; rr < 8; ++rr) {
      int m = (lane < 16) ? rr : rr + 8;
      int gr = rowBlock + wave * 16 + m;
      if (gr < M) {
        float v = acc[rr];
        if (RELU) v = fmaxf(v, 0.0f);
        C[(size_t)gr * NCOL + j * 16 + mrow] = v;
      }
    }
  }
}

// ---------------- WMMA fp32 GEMM, big K: C[M x 128] = A[M x K] * Wt^T + bias ----------------
// Wt is [128 x K] row-major (lin1_w); out[m][n] = sum_k A[m][k]*Wt[n][k]. K multiple of 64.
__global__ __launch_bounds__(256) void k_gemm_bigK(const float* __restrict__ A,
                                                   const float* __restrict__ Wt,
                                                   const float* __restrict__ bias,
                                                   float* __restrict__ C, int M, int K) {
  extern __shared__ float smem[];
  float* ldsW = smem;              // [32][128][2] (k-pair, n) pair-swizzled
  float* ldsA = smem + 64 * 128;   // [128][64]  (m, k)
  const int tid = threadIdx.x, wave = tid >> 5, lane = tid & 31;
  const int rowBlock = blockIdx.x * 128;
  const int mrow = lane & 15, khalf = lane >> 4;
  v8f acc[8] = {};
  for (int kb = 0; kb < K; kb += 64) {
    __syncthreads();
    for (int i = tid; i < 64 * 128; i += 256) {
      int kk = i & 63, n = i >> 6;
      ldsW[(((kk >> 1) * 128 + n) << 1) | (kk & 1)] = Wt[(size_t)n * K + kb + kk];
    }
    for (int i = tid; i < 128 * 64; i += 256) {
      int kk = i & 63, r = i >> 6;
      int gr = rowBlock + r; if (gr >= M) gr = M - 1;
      ldsA[r * 64 + kk] = A[(size_t)gr * K + kb + kk];
    }
    __syncthreads();
    const v2f* ap = (const v2f*)(ldsA + (wave * 16 + mrow) * 64);
    const v2f* wp = (const v2f*)ldsW;
    for (int k = 0; k < 64; k += 4) {
      int p = (k >> 1) + khalf;
      v2f a = ap[p];
#pragma unroll
      for (int j = 0; j < 8; ++j) {
        v2f b = wp[p * 128 + j * 16 + mrow];
        acc[j] = __builtin_amdgcn_wmma_f32_16x16x4_f32(false, a, false, b, (short)0,
                                                       acc[j], false, false);
      }
    }
  }
#pragma unroll
  for (int j = 0; j < 8; ++j)
#pragma unroll
    for (int rr = 0; rr < 8; ++rr) {
      int m = (lane < 16) ? rr : rr + 8;
      int gr = rowBlock + wave * 16 + m;
      if (gr < M) {
        int n = j * 16 + mrow;
        C[(size_t)gr * 128 + n] = acc[j][rr] + bias[n];
      }
    }
}

// ---------------- classifier + log-softmax + NLL loss ----------------
__global__ __launch_bounds__(256) void k_loss(const float* __restrict__ sel2,
                                              const float* __restrict__ cls,
                                              const float* __restrict__ cb,
                                              const int* __restrict__ label,
                                              float* __restrict__ sc, int B) {
  int lane = threadIdx.x & 31, wave = threadIdx.x >> 5;
  int b = blockIdx.x * 8 + wave;
  if (b >= B) return;
  const float* r = sel2 + (size_t)b * 128 + lane * 4;   // d_out+1 is only 4B aligned -> scalar loads
  float s0 = 0.0f, s1 = 0.0f;
#pragma unroll
  for (int i = 0; i < 4; ++i) {
    float v = r[i];
    int d = lane * 4 + i;
    s0 = fmaf(v, cls[d * 2 + 0], s0);
    s1 = fmaf(v, cls[d * 2 + 1], s1);
  }
  s0 = wave_sum32(s0);
  s1 = wave_sum32(s1);
  if (lane == 0) {
    float p0 = s0 + cb[0], p1 = s1 + cb[1];
    float m = fmaxf(p0, p1);
    float lse = m + logf(expf(p0 - m) + expf(p1 - m));
    float pl = ((label[b] == 0) ? p0 : p1) - lse;
    atomicAdd(&sc[3], -pl / (float)B);
  }
}

__global__ void k_finalize(const float* __restrict__ sc, float* __restrict__ out, int B) {
  if (threadIdx.x == 0 && blockIdx.x == 0) {
    out[0] = sc[3];                 // loss
    out[1 + (size_t)B * 128] = sc[0];  // c
  }
}

// ---------------- host orchestration ----------------
extern "C" void kernel_launch(void* const* d_in, const int* in_sizes, int n_in,
                              void* d_out, int out_size, void* d_ws, size_t ws_size,
                              hipStream_t stream) {
  const float* A1     = (const float*)d_in[0];
  const int*   rows   = (const int*)d_in[1];
  const int*   cols   = (const int*)d_in[2];
  const float* vals   = (const float*)d_in[3];
  const int*   bidx   = (const int*)d_in[4];   // (B,1,L) flat
  const int*   blab   = (const int*)d_in[5];
  const float* raw_c  = (const float*)d_in[6];
  const float* nvec   = (const float*)d_in[7];
  const float* Lin1   = (const float*)d_in[8];
  const float* Lin1b  = (const float*)d_in[9];
  const float* gc1w   = (const float*)d_in[10];
  const float* gc1b   = (const float*)d_in[11];
  const float* gc2w   = (const float*)d_in[12];
  const float* gc2b   = (const float*)d_in[13];
  const float* weight = (const float*)d_in[14];  // 128 x 64
  const float* lin1w  = (const float*)d_in[15];  // 128 x (L+1)*64
  const float* lin1b  = (const float*)d_in[16];
  const float* cls    = (const float*)d_in[17];  // 128 x 2
  const float* clsb   = (const float*)d_in[18];

  const int N = in_sizes[7];          // n is (N,1)
  const int E = in_sizes[1];
  const int B = in_sizes[5];
  const int L = in_sizes[4] / B;      // 50
  const int D = 128;
  const size_t ND = (size_t)N * D;

  float* sc   = (float*)d_ws;         // [16] scalars: c, sqrtK, K, loss
  float* ub1  = sc + 16;              // 128
  float* ubg1 = ub1 + 128;            // 128
  float* ubg2 = ubg1 + 128;           // 128
  float* base = sc + 1024;
  float* buf0 = base;                 // t0 / x2
  float* buf1 = base + ND;            // evolving point rows (a1 / x1)
  float* buf2 = base + 2 * ND;        // a2 / gc2 intermediates
  float* buf3 = base + 3 * ND;        // tangents g
  float* buf4 = base + 4 * ND;        // scatter accumulator
  float* SELC = buf1;                 // B*(L+1)*128  (fits in buf1..buf3 region)
  float* SELR = buf3 + (2u << 20);    // B*(L+1)*64   (within buf3/buf4 region)
  float* sel2 = (float*)d_out + 1;    // B*128 output rows

  dim3 blk(256);
  const int gN   = (N + 7) / 8;
  const int gGem = (N + 127) / 128;
  const int gE   = (E + 7) / 8;
  const int gBL  = (B * L + 7) / 8;
  const int gB   = (B + 7) / 8;
  const int gSel = (B * (L + 1) + 127) / 128;
  const int gBig = (B + 127) / 128;
  const size_t shW128 = (size_t)(128 * 128 + 128 * 128) * 4;  // 128 KB
  const size_t shW64  = (size_t)(128 * 64 + 128 * 128) * 4;   //  96 KB
  const size_t shBig  = (size_t)(64 * 128 + 128 * 64) * 4;    //  64 KB
  const int Kbig = (L + 1) * 64;

  k_scalars<<<1, 32, 0, stream>>>(raw_c, sc);
  k_bias_prep<<<1, 32, 0, stream>>>(Lin1b, ub1, sc);
  k_bias_prep<<<1, 32, 0, stream>>>(gc1b, ubg1, sc);
  k_bias_prep<<<1, 32, 0, stream>>>(gc2b, ubg2, sc);

  // a1 = mobius_add(mobius_matvec1(Lin1, A1), bias); a2 = matvec0(n, a1)
  k_logmap0_rows<<<gN, blk, 0, stream>>>(A1, buf0, sc, N);                  // t0
  k_gemm128<128, false><<<gGem, blk, shW128, stream>>>(buf0, Lin1, buf1, N);
  k_expmap0_rows<<<gN, blk, 0, stream>>>(buf1, buf1, sc, N);
  k_madd_bias<<<gN, blk, 0, stream>>>(buf1, ub1, sc, N);
  k_matvec0<<<gN, blk, 0, stream>>>(buf1, nvec, buf2, 0, sc, N);            // a2

  // x_1 = gcn_layer(A1, gc1)
  k_gemm128<128, false><<<gGem, blk, shW128, stream>>>(buf0, gc1w, buf1, N);
  k_expmap0_rows<<<gN, blk, 0, stream>>>(buf1, buf1, sc, N);
  k_madd_bias<<<gN, blk, 0, stream>>>(buf1, ubg1, sc, N);
  k_logmap0_rows<<<gN, blk, 0, stream>>>(buf1, buf3, sc, N);                // g
  k_zero<<<2048, blk, 0, stream>>>(buf4, (long)ND);
  k_scatter<<<gE, blk, 0, stream>>>(rows, cols, vals, buf3, buf4, E);
  k_expmap0_rows<<<gN, blk, 0, stream>>>(buf4, buf1, sc, N);                // x1
  k_matvec0<<<gN, blk, 0, stream>>>(buf1, nvec, buf1, 1, sc, N);            // (1-n) scale
  k_madd0<<<gN, blk, 0, stream>>>(buf1, buf2, sc, N);                       // + a2

  // x_2 = gcn_layer(x_1, gc2)
  k_logmap0_rows<<<gN, blk, 0, stream>>>(buf1, buf0, sc, N);
  k_gemm128<128, false><<<gGem, blk, shW128, stream>>>(buf0, gc2w, buf2, N);
  k_expmap0_rows<<<gN, blk, 0, stream>>>(buf2, buf2, sc, N);
  k_madd_bias<<<gN, blk, 0, stream>>>(buf2, ubg2, sc, N);
  k_logmap0_rows<<<gN, blk, 0, stream>>>(buf2, buf3, sc, N);
  k_zero<<<2048, blk, 0, stream>>>(buf4, (long)ND);
  k_scatter<<<gE, blk, 0, stream>>>(rows, cols, vals, buf3, buf4, E);
  k_expmap0_rows<<<gN, blk, 0, stream>>>(buf4, buf0, sc, N);                // x2 -> buf0

  // head
  k_gather_logmap<<<gBL, blk, 0, stream>>>(buf0, bidx, SELC, sc, B * L, L);
  k_batch_sum<<<gB, blk, 0, stream>>>(SELC, B, L);
  k_gemm128<64, true><<<gSel, blk, shW64, stream>>>(SELC, weight, SELR, B * (L + 1));
  k_gemm_bigK<<<gBig, blk, shBig, stream>>>(SELR, lin1w, lin1b, sel2, B, Kbig);
  k_loss<<<gB, blk, 0, stream>>>(sel2, cls, clsb, blab, sc, B);
  k_finalize<<<1, 32, 0, stream>>>(sc, (float*)d_out, B);

  (void)n_in; (void)out_size; (void)ws_size;
}